// DirectMaskedProjection_9285719294012
// MI455X (gfx1250) — compile-verified
//
#include <hip/hip_runtime.h>
#include <hip/hip_bf16.h>

typedef __attribute__((ext_vector_type(2))) float v2f;
typedef __attribute__((ext_vector_type(8))) float v8f;

#define IMG_H 128
#define IMG_W 128
#define NDEPTH 64
#define HID 64
#define MDZ 128
#define MDY 128
#define MDX 128

// One wave (32 lanes) per pixel. 4 depth tiles of 16 points; hidden layer via
// V_WMMA_F32_16X16X4_F32 with bias folded into K=3 (homogeneous 1.0).
__global__ __launch_bounds__(256) void dmp_wmma_kernel(
    const float* __restrict__ T,    // [1,4,4] row-major
    const float* __restrict__ vol,  // [128,128,128] mask volume
    const float* __restrict__ W1,   // [3,64]
    const float* __restrict__ B1,   // [64]
    const float* __restrict__ W2,   // [64,1]
    const float* __restrict__ B2,   // [1]
    float* __restrict__ out)        // [1,128,128]
{
  const int lane = threadIdx.x & 31;
  const int wid  = threadIdx.x >> 5;
  const int pix  = blockIdx.x * 8 + wid;      // 0..16383
  const int hh   = pix >> 7;
  const int ww   = pix & 127;

  const float px = (float)hh - 63.5f;         // centered grid, PIXEL=1
  const float py = (float)ww - 63.5f;

  // uniform transform load
  const float t00=T[0],  t01=T[1],  t02=T[2],  t03=T[3];
  const float t10=T[4],  t11=T[5],  t12=T[6],  t13=T[7];
  const float t20=T[8],  t21=T[9],  t22=T[10], t23=T[11];
  const float t30=T[12], t31=T[13], t32=T[14], t33=T[15];

  const int  m  = lane & 15;                  // point-in-tile this lane represents
  const bool hi = (lane >= 16);

  // B matrix (4x16) per channel block: rows = [W1(0); W1(1); W1(2); b1]
  // VGPR0 holds K=0 (lanes 0-15) | K=2 (lanes 16-31); VGPR1 holds K=1 | K=3.
  v2f   bmat[4];
  float w2c[4];
#pragma unroll
  for (int blk = 0; blk < 4; ++blk) {
    const int col = m + 16 * blk;
    bmat[blk].x = hi ? W1[2 * HID + col] : W1[0 * HID + col];
    bmat[blk].y = hi ? B1[col]           : W1[1 * HID + col];
    w2c[blk]    = W2[col];
  }

  const float inv_maxabs = 1.0f / 63.5f;      // max |base coord|

  float acc  = 0.0f;                          // sum relu(h)*W2*mask over all elems
  float msum = 0.0f;                          // total masked-point count

  for (int z0 = 0; z0 < NDEPTH; z0 += 16) {
    // ---- transform my point (depth z0+m) ----
    const float pz  = ((float)(z0 + m) - 31.5f) * 2.0f;   // DZ = 2
    const float q0  = t00 * px + t01 * py + t02 * pz + t03;
    const float q1  = t10 * px + t11 * py + t12 * pz + t13;
    const float q2  = t20 * px + t21 * py + t22 * pz + t23;
    const float q3w = t30 * px + t31 * py + t32 * pz + t33;

    // ---- trilinear mask sample (grid_sample, zeros pad, align_corners) ----
    const float xs = (q0 * inv_maxabs + 1.0f) * 0.5f * (float)(MDX - 1);
    const float ys = (q1 * inv_maxabs + 1.0f) * 0.5f * (float)(MDY - 1);
    const float zs = (q2 * inv_maxabs + 1.0f) * 0.5f * (float)(MDZ - 1);
    const float xf = floorf(xs), yf = floorf(ys), zf = floorf(zs);
    const float fx = xs - xf,  fy = ys - yf,  fz = zs - zf;
    const int x0i = (int)xf, y0i = (int)yf, z0i = (int)zf;

    float interp = 0.0f;
#pragma unroll
    for (int dz = 0; dz < 2; ++dz)
#pragma unroll
      for (int dy = 0; dy < 2; ++dy)
#pragma unroll
        for (int dx = 0; dx < 2; ++dx) {
          const int xi = x0i + dx, yi = y0i + dy, zi = z0i + dz;
          const bool valid = (xi >= 0) & (xi < MDX) & (yi >= 0) & (yi < MDY) &
                             (zi >= 0) & (zi < MDZ);
          const float w = (dx ? fx : 1.0f - fx) * (dy ? fy : 1.0f - fy) *
                          (dz ? fz : 1.0f - fz);
          float v = 0.0f;
          if (valid) v = vol[(zi * MDY + yi) * MDX + xi];
          interp += w * v;
        }
    const float maskv = (interp != 0.0f) ? 1.0f : 0.0f;
    msum += hi ? 0.0f : maskv;                // count each point once

    // masks for my 8 C-matrix rows (rows m=0..7 low half, 8..15 high half)
    const int mb = hi ? 8 : 0;
    float mrow[8];
#pragma unroll
    for (int r = 0; r < 8; ++r) mrow[r] = __shfl(maskv, mb + r, 32);

    // ---- A matrix (16x4 f32): K0..2 = q/qw, K3 = 1.0 ----
    const float invw = 1.0f / q3w;
    v2f a;
    a.x = hi ? (q2 * invw) : (q0 * invw);
    a.y = hi ? 1.0f        : (q1 * invw);

    // ---- hidden layer via WMMA, then relu * W2 * mask accumulation ----
#pragma unroll
    for (int blk = 0; blk < 4; ++blk) {
      v8f c = {0.f, 0.f, 0.f, 0.f, 0.f, 0.f, 0.f, 0.f};
      c = __builtin_amdgcn_wmma_f32_16x16x4_f32(
          /*neg_a=*/false, a, /*neg_b=*/false, bmat[blk],
          /*c_mod=*/(short)0, c, /*reuse_a=*/false, /*reuse_b=*/false);
      const float w2 = w2c[blk];
#pragma unroll
      for (int r = 0; r < 8; ++r) {
        const float hv = c[r] > 0.0f ? c[r] : 0.0f;
        acc += hv * mrow[r] * w2;
      }
    }
  }

  // ---- full-wave reduction ----
#pragma unroll
  for (int off = 16; off >= 1; off >>= 1) {
    acc  += __shfl_xor(acc, off, 32);
    msum += __shfl_xor(msum, off, 32);
  }
  if (lane == 0) out[pix] = (acc + B2[0] * msum) * 2.0f;   // * DZ
}

extern "C" void kernel_launch(void* const* d_in, const int* in_sizes, int n_in,
                              void* d_out, int out_size, void* d_ws, size_t ws_size,
                              hipStream_t stream) {
  (void)in_sizes; (void)n_in; (void)out_size; (void)d_ws; (void)ws_size;
  const float* T   = (const float*)d_in[0];
  const float* vol = (const float*)d_in[1];
  const float* W1  = (const float*)d_in[2];
  const float* B1  = (const float*)d_in[3];
  const float* W2  = (const float*)d_in[4];
  const float* B2  = (const float*)d_in[5];
  float* out = (float*)d_out;

  const int npix   = IMG_H * IMG_W;   // 16384 pixels, 1 wave each
  const int blocks = npix / 8;        // 8 waves per 256-thread block
  dmp_wmma_kernel<<<blocks, 256, 0, stream>>>(T, vol, W1, B1, W2, B2, out);
}